// QLSTM_65481071407365
// MI455X (gfx1250) — compile-verified
//
#include <hip/hip_runtime.h>
#include <hip/hip_bf16.h>
#include <math.h>

// QLSTM collapses analytically (see analysis): circuit(a,p) == cos(a[0]+p[0]),
// gates are per-sample scalars, h/c are rank-1 across the hidden dim.
// Phase 1: FP32 WMMA GEMM  (65536 x 128) @ (128 x 4-padded-to-16)
// Phase 2: scalar LSTM recurrence, 256 lanes (one per batch sample)
// Phase 3: rank-1 broadcast of outputs (memory-bound, ~34 MB)

typedef __attribute__((ext_vector_type(2))) float v2f;
typedef __attribute__((ext_vector_type(8))) float v8f;

#define T_STEPS 256
#define BATCH   256
#define DIN     128
#define HID     128
#define ROWS    (T_STEPS * BATCH)   // 65536
#define SW_PITCH 132                // 128 + pad: avoid 16-way LDS bank conflicts

// ---------------- Phase 1: WMMA GEMM ----------------
// grid: ROWS/128 blocks of 256 threads (8 waves); each wave -> one 16-row tile.
__global__ __launch_bounds__(256) void qlstm_gemm_wmma(
    const float* __restrict__ X,   // (ROWS, 128) row-major
    const float* __restrict__ Wf, const float* __restrict__ Wi,
    const float* __restrict__ Wu, const float* __restrict__ Wo,
    float* __restrict__ Apre)      // (ROWS, 4): x-part pre-activations per gate
{
    __shared__ float sW[16 * SW_PITCH];   // 16 "gate columns" (4 real + 12 zero)
    const int tid = threadIdx.x;

    for (int idx = tid; idx < 16 * 128; idx += 256) {
        int g = idx >> 7, k = idx & 127;
        const float* w = (g == 0) ? Wf : (g == 1) ? Wi : (g == 2) ? Wu : Wo;
        sW[g * SW_PITCH + k] = (g < 4) ? w[k] : 0.0f;   // row 0, x-half of W_g
    }
    __syncthreads();

    const int wave = tid >> 5;
    const int lane = tid & 31;
    const int m    = lane & 15;      // M (A row) for lanes; N (B col) as well
    const int hi   = lane >> 4;      // lane half selects K offset {0,2}
    const int row0 = blockIdx.x * 128 + wave * 16;

    const float* arow = X + (size_t)(row0 + m) * DIN;
    const float* brow = &sW[m * SW_PITCH];

    v8f acc = {};
    #pragma unroll
    for (int kc = 0; kc < 32; ++kc) {
        const int k = kc * 4 + 2 * hi;
        // A 16x4 f32 fragment: VGPR0 = K{0|2}, VGPR1 = K{1|3}  (ISA 7.12.2)
        v2f a; a.x = arow[k]; a.y = arow[k + 1];
        // B 4x16 f32 fragment: symmetric layout, lane -> N, half -> K offset
        v2f b; b.x = brow[k]; b.y = brow[k + 1];
        acc = __builtin_amdgcn_wmma_f32_16x16x4_f32(
            /*neg_a=*/false, a, /*neg_b=*/false, b,
            /*c_mod=*/(short)0, acc, /*reuse_a=*/false, /*reuse_b=*/false);
    }

    // C/D layout: VGPR v -> M = v + 8*hi, lane%16 -> N. Only N=0..3 are real gates.
    if (m < 4) {
        const int rbase = row0 + 8 * hi;
        #pragma unroll
        for (int v = 0; v < 8; ++v)
            Apre[(size_t)(rbase + v) * 4 + m] = acc[v];
    }
}

// ---------------- Phase 2: scalar recurrence ----------------
__device__ __forceinline__ float sigmoidf_(float x) { return 1.0f / (1.0f + expf(-x)); }

__global__ __launch_bounds__(256) void qlstm_recur(
    const float* __restrict__ Apre,
    const float* __restrict__ Wf, const float* __restrict__ bf, const float* __restrict__ pf,
    const float* __restrict__ Wi, const float* __restrict__ bi, const float* __restrict__ pi,
    const float* __restrict__ Wu, const float* __restrict__ bu, const float* __restrict__ pu,
    const float* __restrict__ Wo, const float* __restrict__ bo, const float* __restrict__ po,
    float* __restrict__ Hs,    // (T, B) per-sample h scalars
    float* __restrict__ Cfin)  // (B,)  final c scalars
{
    __shared__ float s_hsum[4], s_bp[4];
    const int b = threadIdx.x;

    if (b < 4) {
        const float* w  = (b == 0) ? Wf : (b == 1) ? Wi : (b == 2) ? Wu : Wo;
        const float* bb = (b == 0) ? bf : (b == 1) ? bi : (b == 2) ? bu : bo;
        const float* pp = (b == 0) ? pf : (b == 1) ? pi : (b == 2) ? pu : po;
        float s = 0.0f;
        for (int k = 0; k < HID; ++k) s += w[DIN + k];   // h is rank-1 => W[0,128:].sum()
        s_hsum[b] = s;
        s_bp[b]   = bb[0] + pp[0];
    }
    __syncthreads();

    const float hs0 = s_hsum[0], hs1 = s_hsum[1], hs2 = s_hsum[2], hs3 = s_hsum[3];
    const float bp0 = s_bp[0],   bp1 = s_bp[1],   bp2 = s_bp[2],   bp3 = s_bp[3];

    float c = 0.0f, h = 0.0f;
    for (int t = 0; t < T_STEPS; ++t) {
        const float4 a4 = *(const float4*)&Apre[((size_t)t * BATCH + b) * 4];
        const float f  = sigmoidf_(cosf(a4.x + h * hs0 + bp0));
        const float ig = sigmoidf_(cosf(a4.y + h * hs1 + bp1));
        const float g  = tanhf    (cosf(a4.z + h * hs2 + bp2));
        const float o  = sigmoidf_(cosf(a4.w + h * hs3 + bp3));
        c = f * c + ig * g;
        h = o * tanhf(c);
        Hs[t * BATCH + b] = h;
    }
    Cfin[b] = c;
}

// ---------------- Phase 3: rank-1 broadcast of outputs ----------------
__global__ __launch_bounds__(256) void qlstm_expand(
    const float* __restrict__ Hs, const float* __restrict__ Cfin,
    float4* __restrict__ out)
{
    const int TBH4 = ROWS * (HID / 4);    // outs region in float4s
    const int BH4  = BATCH * (HID / 4);   // hx / cx regions
    const int idx  = blockIdx.x * 256 + threadIdx.x;
    float v;
    if (idx < TBH4) {
        v = Hs[idx >> 5];                                   // (t*B+b) = idx / 32
    } else if (idx < TBH4 + BH4) {
        v = Hs[(T_STEPS - 1) * BATCH + ((idx - TBH4) >> 5)]; // hx = last h
    } else if (idx < TBH4 + 2 * BH4) {
        v = Cfin[(idx - TBH4 - BH4) >> 5];                   // cx
    } else {
        return;
    }
    out[idx] = make_float4(v, v, v, v);
}

extern "C" void kernel_launch(void* const* d_in, const int* in_sizes, int n_in,
                              void* d_out, int out_size, void* d_ws, size_t ws_size,
                              hipStream_t stream)
{
    // setup_inputs() dict order:
    const float* X  = (const float*)d_in[0];
    const float* Wf = (const float*)d_in[1];
    const float* bf = (const float*)d_in[2];
    const float* pf = (const float*)d_in[3];
    const float* Wi = (const float*)d_in[4];
    const float* bi = (const float*)d_in[5];
    const float* pi = (const float*)d_in[6];
    const float* Wu = (const float*)d_in[7];
    const float* bu = (const float*)d_in[8];
    const float* pu = (const float*)d_in[9];
    const float* Wo = (const float*)d_in[10];
    const float* bo = (const float*)d_in[11];
    const float* po = (const float*)d_in[12];

    float* Apre = (float*)d_ws;                 // ROWS*4 floats = 1 MiB
    float* Hs   = Apre + (size_t)ROWS * 4;      // ROWS floats   = 256 KiB
    float* Cfin = Hs + ROWS;                    // BATCH floats

    qlstm_gemm_wmma<<<ROWS / 128, 256, 0, stream>>>(X, Wf, Wi, Wu, Wo, Apre);

    qlstm_recur<<<1, 256, 0, stream>>>(Apre,
                                       Wf, bf, pf, Wi, bi, pi,
                                       Wu, bu, pu, Wo, bo, po,
                                       Hs, Cfin);

    const int total4 = ROWS * (HID / 4) + 2 * BATCH * (HID / 4);
    qlstm_expand<<<(total4 + 255) / 256, 256, 0, stream>>>(Hs, Cfin, (float4*)d_out);
}